// GCN_3702261809847
// MI455X (gfx1250) — compile-verified
//
#include <hip/hip_runtime.h>
#include <hip/hip_bf16.h>

#define DIM 128
#define TM  128   // rows of X per block
#define LDP (DIM + 4)   // padded LDS row stride (528 B = 33*16 -> 16B-aligned rows,
                        // and 132 % 64 banks = 4 -> conflict-free column reads)

typedef __attribute__((ext_vector_type(2))) float v2f;
typedef __attribute__((ext_vector_type(8))) float v8f;

// ---------------------------------------------------------------------------
// gfx1250 async global->LDS copy (ASYNCcnt-tracked), per-lane 16B.
__device__ __forceinline__ unsigned lds_off(const void* p) {
    // generic -> LDS(addrspace 3) -> 32-bit LDS byte offset
    return (unsigned)(unsigned long long)(__attribute__((address_space(3))) const void*)p;
}
__device__ __forceinline__ void async_copy_b128(unsigned lds_byte_addr, const void* gptr) {
    asm volatile("global_load_async_to_lds_b128 %0, %1, off"
                 :: "v"(lds_byte_addr), "v"(gptr) : "memory");
}
__device__ __forceinline__ void wait_asynccnt0() {
    asm volatile("s_wait_asynccnt 0x0" ::: "memory");
}

// ---------------------------------------------------------------------------
// deg[i] = 1.0 (self loop)
__global__ void gcn_deg_init(float* deg, int n) {
    int i = blockIdx.x * blockDim.x + threadIdx.x;
    if (i < n) deg[i] = 1.0f;
}

// deg[dst[e]] += 1
__global__ void gcn_deg_scatter(const int* __restrict__ dst, float* deg, int nE) {
    int e = blockIdx.x * blockDim.x + threadIdx.x;
    if (e < nE) unsafeAtomicAdd(&deg[dst[e]], 1.0f);
}

// dis[i] = rsqrt(deg[i])  (in place)
__global__ void gcn_rsqrt(float* d, int n) {
    int i = blockIdx.x * blockDim.x + threadIdx.x;
    if (i < n) d[i] = __frsqrt_rn(d[i]);
}

// x = max(x, 0) in place
__global__ void gcn_relu(float* x, size_t total) {
    size_t idx = (size_t)blockIdx.x * blockDim.x + threadIdx.x;
    if (idx < total) x[idx] = fmaxf(x[idx], 0.f);
}

// ---------------------------------------------------------------------------
// H = X @ W  via V_WMMA_F32_16X16X4_F32.
// block = 256 threads (8 wave32). X tile (128x128) async-copied to LDS,
// W staged transposed in LDS (Bt[n][k]) so A and B fragments are single
// aligned ds_load_b64 with immediate offsets. Wave w owns M-tile rows
// [16w,16w+16); N-tiles processed 2-wide sharing the A fragment.
__global__ __launch_bounds__(256)
void gcn_gemm_wmma(const float* __restrict__ X, const float* __restrict__ W,
                   float* __restrict__ H, int nrows) {
    __shared__ float Bt[DIM][LDP];   // W transposed: Bt[n][k] = W[k][n]
    __shared__ float Al[TM][LDP];    // X tile, row-major

    const int tid  = threadIdx.x;
    const int row0 = blockIdx.x * TM;

    // ---- stage X tile: per-lane 16B async global->LDS copies -------------
    for (int i = tid; i < TM * (DIM / 4); i += 256) {
        int r  = i >> 5;
        int c4 = (i & 31) << 2;
        int gr = row0 + r;
        if (gr < nrows) {
            async_copy_b128(lds_off(&Al[r][c4]), X + (size_t)gr * DIM + c4);
        } else {
            *(float4*)&Al[r][c4] = make_float4(0.f, 0.f, 0.f, 0.f);  // zero-pad tail
        }
    }
    // ---- stage W transposed (64KB, once per block) -----------------------
    for (int i = tid; i < DIM * (DIM / 4); i += 256) {
        int r  = i >> 5;           // k
        int c4 = (i & 31) << 2;    // n
        float4 v = *(const float4*)(W + (size_t)r * DIM + c4);
        Bt[c4 + 0][r] = v.x;
        Bt[c4 + 1][r] = v.y;
        Bt[c4 + 2][r] = v.z;
        Bt[c4 + 3][r] = v.w;
    }
    wait_asynccnt0();
    __syncthreads();

    const int wv   = tid >> 5;
    const int lane = tid & 31;
    const int m    = lane & 15;   // M (A rows) / N (B cols, C cols) position
    const int half = lane >> 4;   // lanes 16-31: K += 2 (A,B), M += 8 (C/D)

    const float* ap = &Al[wv * 16 + m][half * 2];

    for (int nt = 0; nt < 8; nt += 2) {
        const float* bp0 = &Bt[nt * 16 + m][half * 2];
        const float* bp1 = &Bt[nt * 16 + 16 + m][half * 2];
        v8f c0 = {}, c1 = {};
#pragma unroll
        for (int k = 0; k < DIM; k += 4) {
            v2f a  = *(const v2f*)(ap  + k);
            v2f b0 = *(const v2f*)(bp0 + k);
            v2f b1 = *(const v2f*)(bp1 + k);
            c0 = __builtin_amdgcn_wmma_f32_16x16x4_f32(
                     false, a, false, b0, (short)0, c0, false, false);
            c1 = __builtin_amdgcn_wmma_f32_16x16x4_f32(
                     false, a, false, b1, (short)0, c1, false, false);
        }
#pragma unroll
        for (int r = 0; r < 8; ++r) {
            int gr = row0 + wv * 16 + r + half * 8;   // C layout: M = r + 8*half
            if (gr < nrows) {
                H[(size_t)gr * DIM + nt * 16 + m]      = c0[r];
                H[(size_t)gr * DIM + nt * 16 + 16 + m] = c1[r];
            }
        }
    }
}

// ---------------------------------------------------------------------------
// agg[i][d] = h[i][d] * dis[i]^2 + b[d]   (fully initializes output)
__global__ void gcn_self_init(const float* __restrict__ h, const float* __restrict__ dis,
                              const float* __restrict__ b, float* __restrict__ agg,
                              size_t total) {
    size_t idx = (size_t)blockIdx.x * blockDim.x + threadIdx.x;
    if (idx >= total) return;
    int i = (int)(idx >> 7);
    int d = (int)(idx & (DIM - 1));
    float s = dis[i];
    agg[idx] = h[idx] * s * s + b[d];
}

// one wave per edge: agg[dst] += h[src] * dis[src]*dis[dst]
// (h rows are 512B-contiguous per wave; agg arrays are L2-resident -> fast f32 atomics)
__global__ __launch_bounds__(256)
void gcn_edge_scatter(const float* __restrict__ h, const float* __restrict__ dis,
                      const int* __restrict__ src, const int* __restrict__ dst,
                      float* __restrict__ agg, int nE) {
    const int wv   = threadIdx.x >> 5;
    const int lane = threadIdx.x & 31;
    const int e    = blockIdx.x * 8 + wv;
    if (e >= nE) return;
    const int s = src[e];
    const int t = dst[e];
    const float w = dis[s] * dis[t];
    const float4 v = *(const float4*)(h + (size_t)s * DIM + lane * 4);
    float* p = agg + (size_t)t * DIM + lane * 4;
    unsafeAtomicAdd(p + 0, v.x * w);
    unsafeAtomicAdd(p + 1, v.y * w);
    unsafeAtomicAdd(p + 2, v.z * w);
    unsafeAtomicAdd(p + 3, v.w * w);
}

// ---------------------------------------------------------------------------
extern "C" void kernel_launch(void* const* d_in, const int* in_sizes, int n_in,
                              void* d_out, int out_size, void* d_ws, size_t ws_size,
                              hipStream_t stream) {
    const int*   edges = (const int*)d_in[0];   // [2, E] : row0 = src, row1 = dst
    const float* emb   = (const float*)d_in[1];
    const float* W1    = (const float*)d_in[2];
    const float* b1    = (const float*)d_in[3];
    const float* W2    = (const float*)d_in[4];
    const float* b2    = (const float*)d_in[5];
    float*       out   = (float*)d_out;

    const int nE = in_sizes[0] / 2;
    const int nN = in_sizes[1] / DIM;
    const int* src = edges;
    const int* dst = edges + nE;

    // workspace layout
    char*  ws  = (char*)d_ws;
    float* dis = (float*)ws;                                   // nN floats
    size_t off = (((size_t)nN * sizeof(float)) + 255) & ~(size_t)255;
    float* h   = (float*)(ws + off);                           // nN*128 floats
    float* agg = h + (size_t)nN * DIM;                         // nN*128 floats

    const size_t total = (size_t)nN * DIM;
    const int nThreads = 256;
    const int gN  = (nN + nThreads - 1) / nThreads;
    const int gE  = (nE + nThreads - 1) / nThreads;
    const int gT  = (int)((total + nThreads - 1) / nThreads);
    const int gG  = (nN + TM - 1) / TM;
    const int gS  = (nE + 7) / 8;

    // degree -> dis = rsqrt(deg)
    gcn_deg_init   <<<gN, nThreads, 0, stream>>>(dis, nN);
    gcn_deg_scatter<<<gE, nThreads, 0, stream>>>(dst, dis, nE);
    gcn_rsqrt      <<<gN, nThreads, 0, stream>>>(dis, nN);

    // layer 1: h1 = emb @ W1 ; agg1 = D^-1/2 (A+I) D^-1/2 h1 + b1 ; relu in place
    gcn_gemm_wmma   <<<gG, nThreads, 0, stream>>>(emb, W1, h, nN);
    gcn_self_init   <<<gT, nThreads, 0, stream>>>(h, dis, b1, agg, total);
    gcn_edge_scatter<<<gS, nThreads, 0, stream>>>(h, dis, src, dst, agg, nE);
    gcn_relu        <<<gT, nThreads, 0, stream>>>(agg, total);

    // layer 2: h2 = relu(agg1) @ W2 ; out = norm-agg + b2
    gcn_gemm_wmma   <<<gG, nThreads, 0, stream>>>(agg, W2, h, nN);
    gcn_self_init   <<<gT, nThreads, 0, stream>>>(h, dis, b2, out, total);
    gcn_edge_scatter<<<gS, nThreads, 0, stream>>>(h, dis, src, dst, out, nE);
}